// GlobalAttnLayer_16114717294954
// MI455X (gfx1250) — compile-verified
//
#include <hip/hip_runtime.h>

typedef __attribute__((ext_vector_type(2))) float v2f;
typedef __attribute__((ext_vector_type(8))) float v8f;

#define N_NODES 4096
#define C_CH    128
#define S_SPH   16
#define H_HEADS 8
#define DH      16
#define SC      (S_SPH * C_CH)   // 2048
#define NBATCH  16
#define MASKVAL (-1e30f)

static __device__ __forceinline__ v8f wmma4(v2f a, v2f b, v8f c) {
  // V_WMMA_F32_16X16X4_F32 : D(16x16 f32) = A(16x4 f32) * B(4x16 f32) + C
  return __builtin_amdgcn_wmma_f32_16x16x4_f32(
      /*neg_a=*/false, a, /*neg_b=*/false, b,
      /*c_mod=*/(short)0, c, /*reuse_a=*/false, /*reuse_b=*/false);
}

// ---------------------------------------------------------------------------
// Kernel 0: batch_idx is sorted; compute start[b] = lower_bound(bidx, b) for
// b in [0,16], start[16] = N. One wave, per-lane binary search.
// ---------------------------------------------------------------------------
__global__ __launch_bounds__(32) void batch_starts_kernel(
    const int* __restrict__ bidx, int* __restrict__ start)
{
  const int t = threadIdx.x;
  if (t <= NBATCH) {
    int lo = 0, hi = N_NODES;
    while (lo < hi) {
      const int mid = (lo + hi) >> 1;
      if (bidx[mid] < t) lo = mid + 1; else hi = mid;
    }
    start[t] = lo;
  }
}

// ---------------------------------------------------------------------------
// Kernel 1: q = x_l0 @ wq^T, k = x_l0 @ wk^T, v[:,s,:] = x[:,s,:] @ wv[l(s)]^T (+bv at s=0)
// grid = (N/16, C/16, 2+S), block = 32 (one wave). One 16x16 output tile per wave.
// ---------------------------------------------------------------------------
__global__ __launch_bounds__(32) void so3_qkv_kernel(
    const float* __restrict__ x,  const float* __restrict__ wq,
    const float* __restrict__ wk, const float* __restrict__ wv,
    const float* __restrict__ bv, float* __restrict__ qo,
    float* __restrict__ ko,       float* __restrict__ vo)
{
  const int lane = threadIdx.x & 31;
  const int half = lane >> 4;       // 0: rows/K-lo, 1: rows/K-hi halves
  const int lid  = lane & 15;
  const int rt = blockIdx.x;        // node tile
  const int ct = blockIdx.y;        // out-channel tile
  const int z  = blockIdx.z;        // 0=q, 1=k, 2..17 = v(s=z-2)

  int s_in = 0, is_v = 0;
  const float* W;
  const float* bias = nullptr;
  float* out;
  if (z == 0)      { W = wq; out = qo; }
  else if (z == 1) { W = wk; out = ko; }
  else {
    const int s = z - 2;
    const int l = (s >= 9) ? 3 : ((s >= 4) ? 2 : ((s >= 1) ? 1 : 0));
    s_in = s; is_v = 1;
    W = wv + (size_t)l * C_CH * C_CH;
    out = vo;
    if (s == 0) bias = bv;
  }

  // A: x[node = rt*16+lid][s_in][k]  (lane carries M, regs carry K)
  const float* arow = x + ((size_t)(rt * 16 + lid) * S_SPH + s_in) * C_CH;
  // B: W[col = ct*16+lid][k]         (B[k][n] = W[n][k], lane carries N)
  const float* brow = W + (size_t)(ct * 16 + lid) * C_CH;

  v8f acc = {0.f, 0.f, 0.f, 0.f, 0.f, 0.f, 0.f, 0.f};
#pragma unroll 8
  for (int kk = 0; kk < 32; ++kk) {          // K = 128 = 32 x 4
    const int k0 = kk * 4 + half * 2;
    v2f a, b;
    a.x = arow[k0]; a.y = arow[k0 + 1];
    b.x = brow[k0]; b.y = brow[k0 + 1];
    acc = wmma4(a, b, acc);
  }

  const int col  = ct * 16 + lid;
  const float bz = bias ? bias[col] : 0.f;
#pragma unroll
  for (int r = 0; r < 8; ++r) {
    const int node = rt * 16 + half * 8 + r; // C-layout: row = half*8 + r
    if (is_v)
      out[(size_t)node * SC + (size_t)s_in * C_CH + col] = acc[r] + bz;
    else
      out[(size_t)node * C_CH + col] = acc[r];
  }
}

// ---------------------------------------------------------------------------
// Kernel 2: batch-masked flash attention, block-diagonal skip.
// grid = (N/16 query tiles, H, S/4), block = 32. Each wave: 16 queries x 1 head
// x 4 spherical value slices. batch_idx sorted -> only key tiles overlapping
// [start[b_lo], start[b_hi+1]) can be unmasked; boundary tiles masked per-elem.
// ---------------------------------------------------------------------------
__global__ __launch_bounds__(32) void flash_attn_kernel(
    const float* __restrict__ q, const float* __restrict__ k,
    const float* __restrict__ v, const int* __restrict__ bidx,
    const int* __restrict__ bstart, float* __restrict__ o)
{
  __shared__ float plds[16][17];   // C-layout -> A-layout bounce, +1 pad
  const int lane = threadIdx.x & 31;
  const int half = lane >> 4;
  const int lid  = lane & 15;
  const int qt = blockIdx.x;
  const int h  = blockIdx.y;
  const int sg = blockIdx.z;       // s in [sg*4, sg*4+4)

  // Q tile in A layout: A[m][kd], m = lid, kd = kk*4 + half*2 + {0,1}
  v2f qa[4];
#pragma unroll
  for (int kk = 0; kk < 4; ++kk) {
    const int k0 = kk * 4 + half * 2;
    const float* qp = q + (size_t)(qt * 16 + lid) * C_CH + h * DH + k0;
    qa[kk].x = qp[0]; qa[kk].y = qp[1];
  }

  int bi[8];
#pragma unroll
  for (int r = 0; r < 8; ++r) bi[r] = bidx[qt * 16 + half * 8 + r];

  // admissible key range for this query tile (batch_idx sorted)
  const int b_lo = bidx[qt * 16];
  const int b_hi = bidx[qt * 16 + 15];
  const int kt0  = bstart[b_lo] >> 4;
  const int kt1  = (bstart[b_hi + 1] + 15) >> 4;

  const v8f vzero = {0.f, 0.f, 0.f, 0.f, 0.f, 0.f, 0.f, 0.f};
  float m_i[8], l_i[8];
#pragma unroll
  for (int r = 0; r < 8; ++r) { m_i[r] = MASKVAL; l_i[r] = 0.f; }
  v8f acc[4] = {vzero, vzero, vzero, vzero};

  for (int kt = kt0; kt < kt1; ++kt) {
    const int bj = bidx[kt * 16 + lid];      // lane carries key column n = lid

    // S = Q * K^T  (Dh=16 -> 4 chained WMMAs); B[kd][n] = K[n][kd]
    v8f s = vzero;
#pragma unroll
    for (int kk = 0; kk < 4; ++kk) {
      const int k0 = kk * 4 + half * 2;
      const float* kp = k + (size_t)(kt * 16 + lid) * C_CH + h * DH + k0;
      v2f b; b.x = kp[0]; b.y = kp[1];
      s = wmma4(qa[kk], b, s);
    }

    // scale + batch mask + online softmax (row reductions across 16-lane halves)
    float alpha[8];
#pragma unroll
    for (int r = 0; r < 8; ++r) {
      float sv = (bi[r] == bj) ? s[r] * 0.25f : MASKVAL;  // 1/sqrt(Dh)
      float xm = sv;
      xm = fmaxf(xm, __shfl_xor(xm, 1, 32));
      xm = fmaxf(xm, __shfl_xor(xm, 2, 32));
      xm = fmaxf(xm, __shfl_xor(xm, 4, 32));
      xm = fmaxf(xm, __shfl_xor(xm, 8, 32));
      const float mnew = fmaxf(m_i[r], xm);
      const float a    = __expf(m_i[r] - mnew);  // ==1 while all-masked; later alpha=0 wipes
      const float p    = __expf(sv - mnew);
      float ps = p;
      ps += __shfl_xor(ps, 1, 32);
      ps += __shfl_xor(ps, 2, 32);
      ps += __shfl_xor(ps, 4, 32);
      ps += __shfl_xor(ps, 8, 32);
      l_i[r]  = l_i[r] * a + ps;
      m_i[r]  = mnew;
      alpha[r] = a;
      s[r]    = p;
    }
#pragma unroll
    for (int t = 0; t < 4; ++t)
#pragma unroll
      for (int r = 0; r < 8; ++r) acc[t][r] *= alpha[r];

    // P: C-layout -> A-layout via LDS
    __syncthreads();                              // WAR vs previous iter reads
#pragma unroll
    for (int r = 0; r < 8; ++r) plds[half * 8 + r][lid] = s[r];
    __syncthreads();
    v2f pa[4];
#pragma unroll
    for (int kk = 0; kk < 4; ++kk) {
      const int k0 = kk * 4 + half * 2;
      pa[kk].x = plds[lid][k0]; pa[kk].y = plds[lid][k0 + 1];
    }

    // acc[t] += P(16x16 keys) * V(16 keys x 16 cols) per spherical slice
#pragma unroll
    for (int t = 0; t < 4; ++t) {
      const int sph = sg * 4 + t;
#pragma unroll
      for (int kk = 0; kk < 4; ++kk) {
        const int k0 = kk * 4 + half * 2;
        const float* vp = v + (size_t)(kt * 16 + k0) * SC + sph * C_CH + h * DH + lid;
        v2f b; b.x = vp[0]; b.y = vp[SC];   // V[k0][n], V[k0+1][n]
        acc[t] = wmma4(pa[kk], b, acc[t]);
      }
    }
  }

  // normalize and store (o layout: [n][s][c], c = h*16 + d)
#pragma unroll
  for (int t = 0; t < 4; ++t) {
    const int sph = sg * 4 + t;
#pragma unroll
    for (int r = 0; r < 8; ++r) {
      const int node = qt * 16 + half * 8 + r;
      o[(size_t)node * SC + sph * C_CH + h * DH + lid] = acc[t][r] / l_i[r];
    }
  }
}

// ---------------------------------------------------------------------------
// Kernel 3: y[:,s,:] = o[:,s,:] @ wo[l(s)]^T (+bo at s=0)
// grid = (N/16, C/16, S), block = 32.
// ---------------------------------------------------------------------------
__global__ __launch_bounds__(32) void so3_out_kernel(
    const float* __restrict__ o, const float* __restrict__ wo,
    const float* __restrict__ bo, float* __restrict__ y)
{
  const int lane = threadIdx.x & 31;
  const int half = lane >> 4;
  const int lid  = lane & 15;
  const int rt = blockIdx.x;
  const int ct = blockIdx.y;
  const int s  = blockIdx.z;
  const int l  = (s >= 9) ? 3 : ((s >= 4) ? 2 : ((s >= 1) ? 1 : 0));

  const float* W    = wo + (size_t)l * C_CH * C_CH;
  const float* arow = o + (size_t)(rt * 16 + lid) * SC + s * C_CH;
  const float* brow = W + (size_t)(ct * 16 + lid) * C_CH;

  v8f acc = {0.f, 0.f, 0.f, 0.f, 0.f, 0.f, 0.f, 0.f};
#pragma unroll 8
  for (int kk = 0; kk < 32; ++kk) {
    const int k0 = kk * 4 + half * 2;
    v2f a, b;
    a.x = arow[k0]; a.y = arow[k0 + 1];
    b.x = brow[k0]; b.y = brow[k0 + 1];
    acc = wmma4(a, b, acc);
  }

  const int col  = ct * 16 + lid;
  const float bz = (s == 0) ? bo[col] : 0.f;
#pragma unroll
  for (int r = 0; r < 8; ++r) {
    const int node = rt * 16 + half * 8 + r;
    y[(size_t)node * SC + (size_t)s * C_CH + col] = acc[r] + bz;
  }
}

// ---------------------------------------------------------------------------
extern "C" void kernel_launch(void* const* d_in, const int* in_sizes, int n_in,
                              void* d_out, int out_size, void* d_ws, size_t ws_size,
                              hipStream_t stream) {
  (void)in_sizes; (void)n_in; (void)out_size; (void)ws_size;
  const float* x    = (const float*)d_in[0];
  const int*   bidx = (const int*)  d_in[1];
  const float* wq   = (const float*)d_in[2];
  const float* wk   = (const float*)d_in[3];
  const float* wv   = (const float*)d_in[4];
  const float* bv   = (const float*)d_in[5];
  const float* wo   = (const float*)d_in[6];
  const float* bo   = (const float*)d_in[7];

  // workspace layout (floats): q | k | v | o | bstart(int)
  // = 4096*(128+128+2048+2048)*4B + 128B  ~ 71 MB
  float* q = (float*)d_ws;
  float* k = q + (size_t)N_NODES * C_CH;
  float* v = k + (size_t)N_NODES * C_CH;
  float* o = v + (size_t)N_NODES * SC;
  int* bst = (int*)(o + (size_t)N_NODES * SC);
  float* y = (float*)d_out;

  const dim3 blk(32, 1, 1);
  batch_starts_kernel<<<dim3(1, 1, 1), blk, 0, stream>>>(bidx, bst);
  so3_qkv_kernel  <<<dim3(N_NODES / 16, C_CH / 16, 2 + S_SPH), blk, 0, stream>>>(
      x, wq, wk, wv, bv, q, k, v);
  flash_attn_kernel<<<dim3(N_NODES / 16, H_HEADS, S_SPH / 4),   blk, 0, stream>>>(
      q, k, v, bidx, bst, o);
  so3_out_kernel  <<<dim3(N_NODES / 16, C_CH / 16, S_SPH),      blk, 0, stream>>>(
      o, wo, bo, y);
}